// InductiveRGCN_49959059587594
// MI455X (gfx1250) — compile-verified
//
#include <hip/hip_runtime.h>

typedef __attribute__((ext_vector_type(2))) float v2f;
typedef __attribute__((ext_vector_type(8))) float v8f;

#define TILE_K_MAX 128   // Fin = Fh = 128 in this model

// ---------------------------------------------------------------------------
// C[M,F] = A[M,K] @ B[K,F] (+ bias), fp32 via V_WMMA_F32_16X16X4_F32.
// Block: 128 threads = 4 wave32s; each wave computes one 16x16 output tile,
// so a block covers a 16-row x 64-col strip. Grid: (ceil(M/16), F/64).
// The 16xK A-tile is staged straight into LDS with CDNA5 async copies
// (GLOBAL_LOAD_ASYNC_TO_LDS_B128, tracked on ASYNCcnt) -- no VGPR round trip.
// B (the small weight matrix, <=64KB) streams from global / L2.
// ---------------------------------------------------------------------------
__global__ __launch_bounds__(128)
void gemm_f32_wmma(const float* __restrict__ A, const float* __restrict__ B,
                   const float* __restrict__ bias, float* __restrict__ C,
                   int M, int K, int F) {
  __shared__ float As[16 * TILE_K_MAX];
  const int tid  = threadIdx.x;
  const int lane = tid & 31;
  const int wave = tid >> 5;
  const int row0 = blockIdx.x * 16;
  const int col0 = blockIdx.y * 64 + wave * 16;

  // Async-stage the 16 x K fp32 A-tile into LDS (K is a multiple of 4).
  const int nvec = (16 * K) >> 2;
  for (int i = tid; i < nvec; i += 128) {
    const int flat = i << 2;
    const int r    = flat / K;
    const int k    = flat - r * K;
    const int grow = min(row0 + r, M - 1);   // clamp; M%16==0 in practice
    const float* gsrc = A + (size_t)grow * K + k;
    // Generic pointers into LDS carry the LDS byte offset in their low 32 bits.
    const unsigned ldsaddr = (unsigned)(unsigned long long)(As + flat);
    asm volatile("global_load_async_to_lds_b128 %0, %1, off"
                 :: "v"(ldsaddr), "v"(gsrc) : "memory");
  }
  asm volatile("s_wait_asynccnt 0x0" ::: "memory");
  __syncthreads();

  const int hi = lane >> 4;   // 0 for lanes 0-15, 1 for lanes 16-31
  const int lo = lane & 15;

  v8f c = {};
  for (int k = 0; k < K; k += 4) {
    const int kq = k + 2 * hi;            // ISA A/B 16x4 layout: hi lanes hold K+2
    v2f a, b;
    a.x = As[lo * K + kq];                // A[M=lo][kq]
    a.y = As[lo * K + kq + 1];            // A[M=lo][kq+1]
    const float* Brow = B + (size_t)kq * F + col0 + lo;
    b.x = Brow[0];                        // B[kq][N=lo]
    b.y = Brow[F];                        // B[kq+1][N=lo]
    // 8 args: (neg_a, A, neg_b, B, c_mod, C, reuse_a, reuse_b)
    c = __builtin_amdgcn_wmma_f32_16x16x4_f32(false, a, false, b, (short)0, c,
                                              false, false);
  }

  // C/D layout: VGPR j -> M = j (lanes 0-15) or j+8 (lanes 16-31), N = lane&15.
  const int   n     = col0 + lo;
  const float badd  = bias ? bias[n] : 0.0f;
  const int   mbase = row0 + 8 * hi;
#pragma unroll
  for (int j = 0; j < 8; ++j) {
    const int m = mbase + j;
    if (m < M) C[(size_t)m * F + n] = c[j] + badd;
  }
}

// Per-(dst, relation) in-degree counts.
__global__ void count_kernel(const int* __restrict__ dst,
                             const int* __restrict__ et,
                             unsigned* __restrict__ cnt, int E, int R) {
  int e = blockIdx.x * blockDim.x + threadIdx.x;
  if (e < E) atomicAdd(&cnt[(size_t)dst[e] * R + et[e]], 1u);
}

__global__ void invcnt_kernel(const unsigned* __restrict__ cnt,
                              float* __restrict__ inv, int n) {
  int i = blockIdx.x * blockDim.x + threadIdx.x;
  if (i < n) inv[i] = 1.0f / fmaxf((float)cnt[i], 1.0f);
}

// Wave-per-edge float4 gather + scaled float atomic scatter.
__device__ __forceinline__ void scatter_row(const float* __restrict__ hrow,
                                            float* __restrict__ orow,
                                            float scale, int F, int lane) {
  const float4* h4 = (const float4*)hrow;
  const int nq = F >> 2;
  for (int q = lane; q < nq; q += 32) {
    const float4 v = h4[q];
    float* op = orow + (q << 2);
    atomicAdd(op + 0, v.x * scale);      // global_atomic_add_f32
    atomicAdd(op + 1, v.y * scale);
    atomicAdd(op + 2, v.z * scale);
    atomicAdd(op + 3, v.w * scale);
  }
}

// Thin path: one relation per pass, H holds only that relation's features.
__global__ void aggregate_rel_kernel(const float* __restrict__ H,
                                     const int* __restrict__ src,
                                     const int* __restrict__ dst,
                                     const int* __restrict__ et,
                                     const float* __restrict__ inv,
                                     float* __restrict__ out,
                                     int E, int F, int r, int R) {
  const long long gtid = (long long)blockIdx.x * blockDim.x + threadIdx.x;
  const int e    = (int)(gtid >> 5);
  const int lane = (int)(gtid & 31);
  if (e >= E) return;
  if (et[e] != r) return;                  // uniform across the wave
  const int   s     = src[e];
  const int   d     = dst[e];
  const float scale = inv[(size_t)d * R + r];
  scatter_row(H + (size_t)s * F, out + (size_t)d * F, scale, F, lane);
}

// Fat path: all relations materialized; single pass, no relation predicate.
__global__ void aggregate_all_kernel(const float* __restrict__ Hall,
                                     const int* __restrict__ src,
                                     const int* __restrict__ dst,
                                     const int* __restrict__ et,
                                     const float* __restrict__ inv,
                                     float* __restrict__ out,
                                     int E, int F, int R, size_t relStride) {
  const long long gtid = (long long)blockIdx.x * blockDim.x + threadIdx.x;
  const int e    = (int)(gtid >> 5);
  const int lane = (int)(gtid & 31);
  if (e >= E) return;
  const int   r     = et[e];
  const int   s     = src[e];
  const int   d     = dst[e];
  const float scale = inv[(size_t)d * R + r];
  const float* hrow = Hall + (size_t)r * relStride + (size_t)s * F;
  scatter_row(hrow, out + (size_t)d * F, scale, F, lane);
}

__global__ void relu_kernel(float* __restrict__ x, long long n) {
  long long i = (long long)blockIdx.x * blockDim.x + threadIdx.x;
  if (i < n) x[i] = fmaxf(x[i], 0.0f);
}

// ---------------------------------------------------------------------------
// Host-side orchestration (graph-capture safe: kernels + hipMemsetAsync only).
// ---------------------------------------------------------------------------
extern "C" void kernel_launch(void* const* d_in, const int* in_sizes, int n_in,
                              void* d_out, int out_size, void* d_ws, size_t ws_size,
                              hipStream_t stream) {
  const float* x     = (const float*)d_in[0];
  const int*   eidx  = (const int*)d_in[1];   // [2, E] flat: src then dst
  const int*   etype = (const int*)d_in[2];
  const float* W1    = (const float*)d_in[3];
  const float* root1 = (const float*)d_in[4];
  const float* b1    = (const float*)d_in[5];
  const float* W2    = (const float*)d_in[6];
  const float* root2 = (const float*)d_in[7];
  const float* b2    = (const float*)d_in[8];

  const int Fh   = in_sizes[5];                    // 128
  const int Fin  = in_sizes[4] / Fh;               // 128
  const int N    = in_sizes[0] / Fin;              // 50000
  const int R    = in_sizes[3] / (Fin * Fh);       // 8
  const int Fout = in_sizes[8];                    // 64
  const int E    = in_sizes[2];                    // 800000

  const int* src = eidx;
  const int* dst = eidx + E;

  char*  ws  = (char*)d_ws;
  size_t off = 0;
  auto carve = [&](size_t bytes) -> void* {
    void* p = ws + off;
    off = (off + bytes + 255) & ~(size_t)255;
    return p;
  };
  unsigned* cnt  = (unsigned*)carve((size_t)N * R * sizeof(unsigned));
  float*    inv  = (float*)   carve((size_t)N * R * sizeof(float));
  float*    acc1 = (float*)   carve((size_t)N * Fh * sizeof(float));
  const size_t HbytesFat  = (size_t)R * N * Fh * sizeof(float);   // ~205 MB
  const size_t HbytesThin = (size_t)N * Fh * sizeof(float);       // ~25.6 MB
  const bool fat = (off + HbytesFat) <= ws_size;    // deterministic in ws_size
  float* H = (float*)carve(fat ? HbytesFat : HbytesThin);
  float* out = (float*)d_out;

  // Degree counts (shared by both layers).
  hipMemsetAsync(cnt, 0, (size_t)N * R * sizeof(unsigned), stream);
  count_kernel<<<(E + 255) / 256, 256, 0, stream>>>(dst, etype, cnt, E, R);
  invcnt_kernel<<<(N * R + 255) / 256, 256, 0, stream>>>(cnt, inv, N * R);

  const int aggBlocks = (int)(((long long)E * 32 + 255) / 256);

  // ---- Layer 1: acc1 = relu(x@root1 + b1 + sum_r mean_r(W1_r x_src)) ----
  dim3 g1((N + 15) / 16, Fh / 64);
  gemm_f32_wmma<<<g1, 128, 0, stream>>>(x, root1, b1, acc1, N, Fin, Fh);
  if (fat) {
    for (int r = 0; r < R; ++r)
      gemm_f32_wmma<<<g1, 128, 0, stream>>>(x, W1 + (size_t)r * Fin * Fh,
                                            nullptr, H + (size_t)r * N * Fh,
                                            N, Fin, Fh);
    aggregate_all_kernel<<<aggBlocks, 256, 0, stream>>>(
        H, src, dst, etype, inv, acc1, E, Fh, R, (size_t)N * Fh);
  } else {
    for (int r = 0; r < R; ++r) {
      gemm_f32_wmma<<<g1, 128, 0, stream>>>(x, W1 + (size_t)r * Fin * Fh,
                                            nullptr, H, N, Fin, Fh);
      aggregate_rel_kernel<<<aggBlocks, 256, 0, stream>>>(
          H, src, dst, etype, inv, acc1, E, Fh, r, R);
    }
  }
  relu_kernel<<<(int)(((long long)N * Fh + 255) / 256), 256, 0, stream>>>(
      acc1, (long long)N * Fh);

  // ---- Layer 2: out = acc1@root2 + b2 + sum_r mean_r(W2_r acc1_src) ----
  dim3 g2((N + 15) / 16, Fout / 64);
  gemm_f32_wmma<<<g2, 128, 0, stream>>>(acc1, root2, b2, out, N, Fh, Fout);
  if (fat) {
    for (int r = 0; r < R; ++r)
      gemm_f32_wmma<<<g2, 128, 0, stream>>>(acc1, W2 + (size_t)r * Fh * Fout,
                                            nullptr, H + (size_t)r * N * Fout,
                                            N, Fh, Fout);
    aggregate_all_kernel<<<aggBlocks, 256, 0, stream>>>(
        H, src, dst, etype, inv, out, E, Fout, R, (size_t)N * Fout);
  } else {
    for (int r = 0; r < R; ++r) {
      gemm_f32_wmma<<<g2, 128, 0, stream>>>(acc1, W2 + (size_t)r * Fh * Fout,
                                            nullptr, H, N, Fh, Fout);
      aggregate_rel_kernel<<<aggBlocks, 256, 0, stream>>>(
          H, src, dst, etype, inv, out, E, Fout, r, R);
    }
  }
}